// TopologicalRegularization_81003083202858
// MI455X (gfx1250) — compile-verified
//
#include <hip/hip_runtime.h>

// ---------------------------------------------------------------------------
// TopologicalRegularization: loss = 0.01 * mean(H0 persistences)
//   H0 persistences of Rips filtration == MST edge lengths of the point cloud.
//   Phase 0: sq[i] = |x_i|^2
//   Phase 1: D[i][j] = sqrt(max(sq[i]+sq[j]-2*<x_i,x_j>, 1e-12))  (WMMA fp32)
//   Phase 2: Prim's MST, single persistent workgroup, sum edges -> scalar
// ---------------------------------------------------------------------------

typedef float v2f __attribute__((ext_vector_type(2)));
typedef float v8f __attribute__((ext_vector_type(8)));

#define NPTS 4096
#define DIM  128
#define WEIGHT 0.01f
#define EPSF 1e-12f

// --------------------------- Phase 0: squared norms -------------------------
__global__ void sqnorm_kernel(const float* __restrict__ x, float* __restrict__ sq) {
  const int i = blockIdx.x * blockDim.x + threadIdx.x;  // 0..4095
  const float4* xr = (const float4*)(x + (size_t)i * DIM);
  float s = 0.0f;
#pragma unroll
  for (int k = 0; k < DIM / 4; ++k) {
    float4 v = xr[k];
    s = fmaf(v.x, v.x, s);
    s = fmaf(v.y, v.y, s);
    s = fmaf(v.z, v.z, s);
    s = fmaf(v.w, v.w, s);
  }
  sq[i] = s;
}

// ------------------- Phase 1: distance matrix via fp32 WMMA -----------------
// One wave computes one 16x16 tile of D. grid = (256 tile-cols, 32), 8 waves
// per 256-thread block, wave w handles tile-row blockIdx.y*8+w.
//
// V_WMMA_F32_16X16X4_F32 lane layouts (ISA 7.12.2):
//   A (16x4, MxK): lane L<16 -> M=L,   v0:K=0, v1:K=1 ; lane L>=16 -> M=L-16, v0:K=2, v1:K=3
//   B (4x16, KxN): lane L<16 -> N=L,   v0:K=0, v1:K=1 ; lane L>=16 -> N=L-16, v0:K=2, v1:K=3
//   C/D (16x16):   lane L<16 -> N=L, vgpr r -> M=r ;    lane L>=16 -> N=L-16, vgpr r -> M=8+r
__global__ void __launch_bounds__(256)
dist_kernel(const float* __restrict__ x, const float* __restrict__ sq,
            float* __restrict__ Dmat) {
  const int lane = threadIdx.x & 31;
  const int wave = threadIdx.x >> 5;
  const int ti = blockIdx.y * 8 + wave;  // tile row 0..255
  const int tj = blockIdx.x;             // tile col 0..255
  const int i0 = ti * 16;
  const int j0 = tj * 16;

  const int l15  = lane & 15;
  const int half = lane >> 4;            // selects K-pair {0,1} vs {2,3}
  const size_t arow = (size_t)(i0 + l15) * DIM;
  const size_t brow = (size_t)(j0 + l15) * DIM;

  v8f c = {};
#pragma unroll
  for (int k = 0; k < DIM; k += 4) {
    const int kk = k + half * 2;
    const float2 av = *(const float2*)(x + arow + kk);
    const float2 bv = *(const float2*)(x + brow + kk);
    v2f a, b;
    a[0] = av.x; a[1] = av.y;
    b[0] = bv.x; b[1] = bv.y;
    // 8 args: (neg_a, A, neg_b, B, c_mod, C, reuse_a, reuse_b)
    c = __builtin_amdgcn_wmma_f32_16x16x4_f32(false, a, false, b, (short)0, c,
                                              false, false);
  }

  const int   col = j0 + l15;
  const float sqj = sq[col];
#pragma unroll
  for (int r = 0; r < 8; ++r) {
    const int   row = i0 + half * 8 + r;
    float d2 = sq[row] + sqj - 2.0f * c[r];
    d2 = fmaxf(d2, EPSF);
    Dmat[(size_t)row * NPTS + col] = __builtin_sqrtf(d2);
  }
}

// ----------------------- Phase 2: Prim's MST (serial) -----------------------
// One workgroup, 1024 threads (32 wave32 waves). Thread t owns nodes
// 4t..4t+3; mindist lives in registers. Argmin packs (bits(val)<<32)|idx into
// a u64 (order-preserving for non-negative floats; smallest-index tiebreak).
__global__ void __launch_bounds__(1024)
prim_kernel(const float* __restrict__ Dmat, float* __restrict__ out) {
  __shared__ unsigned long long red[32];
  __shared__ unsigned long long bcast;

  const int t    = threadIdx.x;
  const int lane = t & 31;
  const int wave = t >> 5;
  const int base = t * 4;

  float md[4];
  int alive = 0xF;
  {
    const float4 r0 = *(const float4*)(Dmat + base);  // row of node 0
    md[0] = r0.x; md[1] = r0.y; md[2] = r0.z; md[3] = r0.w;
  }
  if (t == 0) { alive &= ~1; md[0] = __builtin_inff(); }  // node 0 visited

  float edgeSum = 0.0f;

  for (int it = 0; it < NPTS - 1; ++it) {
    // ---- local packed argmin over owned alive nodes ----
    unsigned long long pm = ~0ull;
#pragma unroll
    for (int e = 0; e < 4; ++e) {
      if (alive & (1 << e)) {
        unsigned long long p =
            ((unsigned long long)__float_as_uint(md[e]) << 32) |
            (unsigned)(base + e);
        pm = (p < pm) ? p : pm;
      }
    }
    // ---- wave32 reduction ----
#pragma unroll
    for (int off = 16; off > 0; off >>= 1) {
      unsigned long long o = __shfl_xor(pm, off, 32);
      pm = (o < pm) ? o : pm;
    }
    if (lane == 0) red[wave] = pm;
    __syncthreads();
    // ---- cross-wave reduction in wave 0 ----
    if (wave == 0) {
      unsigned long long q = red[lane];
#pragma unroll
      for (int off = 16; off > 0; off >>= 1) {
        unsigned long long o = __shfl_xor(q, off, 32);
        q = (o < q) ? o : q;
      }
      if (lane == 0) {
        bcast = q;
        edgeSum += __uint_as_float((unsigned)(q >> 32));  // MST edge length
      }
    }
    __syncthreads();
    // ---- relax against row D[j] (coalesced float4, L2-resident) ----
    const unsigned long long q = bcast;
    const int j = (int)(q & 0xFFFFFFFFu);
    if ((j >> 2) == t) {  // newly visited node is one of ours
      const int e = j & 3;
      alive &= ~(1 << e);
      md[e] = __builtin_inff();
    }
    const float4 rv = *(const float4*)(Dmat + (size_t)j * NPTS + base);
    const float rr[4] = {rv.x, rv.y, rv.z, rv.w};
#pragma unroll
    for (int e = 0; e < 4; ++e)
      if (alive & (1 << e)) md[e] = fminf(md[e], rr[e]);
  }

  if (t == 0) out[0] = WEIGHT * (edgeSum / (float)(NPTS - 1));
}

// ---------------------------------------------------------------------------
extern "C" void kernel_launch(void* const* d_in, const int* in_sizes, int n_in,
                              void* d_out, int out_size, void* d_ws,
                              size_t ws_size, hipStream_t stream) {
  (void)in_sizes; (void)n_in; (void)out_size; (void)ws_size;
  const float* x   = (const float*)d_in[0];   // [4096, 128] fp32
  float*       out = (float*)d_out;           // scalar fp32

  // Workspace layout: [ D : 4096*4096 f32 (64MB, L2-resident) ][ sq : 4096 f32 ]
  float* Dmat = (float*)d_ws;
  float* sq   = Dmat + (size_t)NPTS * NPTS;

  sqnorm_kernel<<<NPTS / 256, 256, 0, stream>>>(x, sq);

  dim3 grid(NPTS / 16, NPTS / 16 / 8);  // (256, 32), 8 waves/block
  dist_kernel<<<grid, 256, 0, stream>>>(x, sq, Dmat);

  prim_kernel<<<1, 1024, 0, stream>>>(Dmat, out);
}